// GCNModule_31164282699782
// MI455X (gfx1250) — compile-verified
//
#include <hip/hip_runtime.h>

// GCN 3-layer forward for gfx1250 (MI455X), fp32 throughout.
// Matrix path: V_WMMA_F32_16X16X4_F32 (wave32, 16x16 tile, K-step 4).

#define H_DIM 64
#define EPS_BN 1e-5f
#define BN_BLOCKS 512

typedef float v2f __attribute__((ext_vector_type(2)));
typedef float v8f __attribute__((ext_vector_type(8)));

// ---------------- degree / norm ----------------

__global__ void deg_init_kernel(float* __restrict__ deg, int n) {
  int i = blockIdx.x * blockDim.x + threadIdx.x;
  if (i < n) deg[i] = 1.0f;  // self-loop contributes 1
}

__global__ void deg_count_kernel(const int* __restrict__ dst, float* __restrict__ deg, int e) {
  int i = blockIdx.x * blockDim.x + threadIdx.x;
  if (i < e) unsafeAtomicAdd(&deg[dst[i]], 1.0f);  // global_atomic_add_f32
}

__global__ void dinv_kernel(float* __restrict__ deg, int n) {
  int i = blockIdx.x * blockDim.x + threadIdx.x;
  if (i < n) deg[i] = rsqrtf(deg[i]);  // deg >= 1 always
}

// ---------------- GEMM: H = X @ W via f32 WMMA ----------------
// X: n_rows x K (row-major), W: K x 64 (row-major), Hout: n_rows x 64.
// One wave computes a 16x64 output stripe with 4 accumulators of 16x16.
// W is staged transposed in LDS (Wt[n*K+k]) so each B-fragment is one ds_load_b64.

__global__ void __launch_bounds__(128) gemm_wmma_kernel(
    const float* __restrict__ X, const float* __restrict__ W,
    float* __restrict__ Hout, int n_rows, int K) {
  __shared__ float Wt[128 * H_DIM];  // 32 KB max (K<=128)

  int tid = threadIdx.x;
  int total = K * H_DIM;
  for (int i = tid; i < total; i += blockDim.x) {
    int k = i >> 6;      // row of W
    int n = i & 63;      // col of W
    Wt[n * K + k] = W[i];
  }
  __syncthreads();

  int wave  = tid >> 5;
  int lane  = tid & 31;
  int tile  = blockIdx.x * 4 + wave;
  int ntile = n_rows >> 4;              // n_rows assumed multiple of 16 (100000/16 = 6250)
  if (tile >= ntile) return;            // wave-uniform: EXEC stays all-ones for WMMA

  int m     = lane & 15;                // M for A-frag, N for B/D-frags
  int khalf = lane >> 4;                // selects K pair within the 4-wide K step
  int row0  = tile << 4;

  v8f acc[4];
#pragma unroll
  for (int j = 0; j < 4; ++j)
#pragma unroll
    for (int r = 0; r < 8; ++r) acc[j][r] = 0.0f;

  const float* xrow = X + (size_t)(row0 + m) * K;

  for (int k0 = 0; k0 < K; k0 += 4) {
    int kk = k0 + 2 * khalf;
    // A 16x4 f32 frag: lane m holds X[row0+m][kk], X[row0+m][kk+1]
    v2f a = *(const v2f*)(xrow + kk);
#pragma unroll
    for (int j = 0; j < 4; ++j) {
      // B 4x16 f32 frag: lane m holds W[kk][j*16+m], W[kk+1][j*16+m]
      v2f b = *(const v2f*)(&Wt[(j * 16 + m) * K + kk]);
      acc[j] = __builtin_amdgcn_wmma_f32_16x16x4_f32(
          /*neg_a=*/false, a, /*neg_b=*/false, b,
          /*c_mod=*/(short)0, acc[j], /*reuse_a=*/false, /*reuse_b=*/false);
    }
  }

  // D layout: VGPR r -> M = r (lanes 0-15) or r+8 (lanes 16-31); N = lane&15
#pragma unroll
  for (int r = 0; r < 8; ++r) {
    int row = row0 + r + 8 * khalf;
    float* hr = Hout + (size_t)row * H_DIM;
#pragma unroll
    for (int j = 0; j < 4; ++j) hr[j * 16 + m] = acc[j][r];
  }
}

// ---------------- aggregation: out = segsum(h[src]*norm) + self-loop + bias ----------------

__global__ void aggr_init_kernel(const float* __restrict__ h, const float* __restrict__ dinv,
                                 const float* __restrict__ bias, float* __restrict__ agg,
                                 int n) {
  int i = blockIdx.x * blockDim.x + threadIdx.x;   // n*64 <= 6.4M fits in int
  if (i >= n * H_DIM) return;
  int row = i >> 6;
  int f = i & 63;
  float di = dinv[row];
  agg[i] = h[i] * di * di + bias[f];               // self-loop: norm = dinv^2
}

__global__ void aggr_edges_kernel(const float* __restrict__ h, const float* __restrict__ dinv,
                                  const int* __restrict__ src, const int* __restrict__ dst,
                                  float* __restrict__ agg, int e) {
  long t = (long)blockIdx.x * blockDim.x + threadIdx.x;
  if (t >= (long)e * 16) return;
  int ei = (int)(t >> 4);
  int q  = ((int)t & 15) << 2;                     // 16 lanes cover one 64-float row
  int s = src[ei];
  int d = dst[ei];
  float nr = dinv[s] * dinv[d];
  const float4 hv = *(const float4*)(h + (size_t)s * H_DIM + q);
  float* out = agg + (size_t)d * H_DIM + q;
  unsafeAtomicAdd(out + 0, hv.x * nr);
  unsafeAtomicAdd(out + 1, hv.y * nr);
  unsafeAtomicAdd(out + 2, hv.z * nr);
  unsafeAtomicAdd(out + 3, hv.w * nr);
}

// ---------------- BatchNorm (training stats) + PReLU ----------------

__global__ void __launch_bounds__(256) bn_partial_kernel(const float* __restrict__ x,
                                                         float* __restrict__ partial, int n) {
  int tid = threadIdx.x;
  int f = tid & 63;
  int grp = tid >> 6;  // 4 row-groups per block
  float s = 0.0f, s2 = 0.0f;
  for (int row = blockIdx.x * 4 + grp; row < n; row += gridDim.x * 4) {
    float v = x[(size_t)row * H_DIM + f];
    s += v;
    s2 += v * v;
  }
  __shared__ float rs[256];
  __shared__ float rs2[256];
  rs[tid] = s;
  rs2[tid] = s2;
  __syncthreads();
  if (tid < 64) {
    s  = rs[tid] + rs[tid + 64] + rs[tid + 128] + rs[tid + 192];
    s2 = rs2[tid] + rs2[tid + 64] + rs2[tid + 128] + rs2[tid + 192];
    partial[blockIdx.x * 128 + tid]      = s;
    partial[blockIdx.x * 128 + 64 + tid] = s2;
  }
}

__global__ void __launch_bounds__(128) bn_final_kernel(const float* __restrict__ partial,
                                                       int nblocks,
                                                       const float* __restrict__ g,
                                                       const float* __restrict__ be,
                                                       float* __restrict__ ss, float inv_n) {
  int t = threadIdx.x;  // 128 threads: 64 sums + 64 sumsq columns
  float acc = 0.0f;
  for (int b = 0; b < nblocks; ++b) acc += partial[b * 128 + t];
  __shared__ float tot[128];
  tot[t] = acc;
  __syncthreads();
  if (t < 64) {
    float mu  = tot[t] * inv_n;
    float var = tot[64 + t] * inv_n - mu * mu;   // biased variance
    float sc  = g[t] * rsqrtf(var + EPS_BN);
    ss[t]      = sc;
    ss[64 + t] = be[t] - mu * sc;
  }
}

__global__ void bn_apply_prelu_kernel(const float* __restrict__ x, const float* __restrict__ ss,
                                      const float* __restrict__ a, float* __restrict__ y, int n) {
  int i = blockIdx.x * blockDim.x + threadIdx.x;
  if (i >= n * H_DIM) return;
  int f = i & 63;
  float v = x[i] * ss[f] + ss[64 + f];
  y[i] = (v >= 0.0f) ? v : a[f] * v;
}

// ---------------- host-side orchestration ----------------

extern "C" void kernel_launch(void* const* d_in, const int* in_sizes, int n_in,
                              void* d_out, int out_size, void* d_ws, size_t ws_size,
                              hipStream_t stream) {
  const float* x   = (const float*)d_in[0];
  const int*   ei  = (const int*)d_in[1];
  const float* W1  = (const float*)d_in[2];
  const float* b1  = (const float*)d_in[3];
  const float* g1  = (const float*)d_in[4];
  const float* be1 = (const float*)d_in[5];
  const float* a1  = (const float*)d_in[6];
  const float* W2  = (const float*)d_in[7];
  const float* b2  = (const float*)d_in[8];
  const float* g2  = (const float*)d_in[9];
  const float* be2 = (const float*)d_in[10];
  const float* a2  = (const float*)d_in[11];
  const float* W3  = (const float*)d_in[12];
  const float* b3  = (const float*)d_in[13];
  const float* g3  = (const float*)d_in[14];
  const float* be3 = (const float*)d_in[15];
  const float* a3  = (const float*)d_in[16];

  const int N = in_sizes[0] / 128;   // 100000
  const int E = in_sizes[1] / 2;     // 1600000
  const int* src = ei;
  const int* dst = ei + E;

  float* ws = (float*)d_ws;
  float* dinv    = ws;  ws += N;
  float* B0      = ws;  ws += (size_t)N * H_DIM;   // GEMM output h
  float* B1      = ws;  ws += (size_t)N * H_DIM;   // aggregate / activation (in-place apply)
  float* partial = ws;  ws += (size_t)BN_BLOCKS * 128;
  float* ssbuf   = ws;  ws += 128;

  const int T = 256;
  deg_init_kernel<<<(N + T - 1) / T, T, 0, stream>>>(dinv, N);
  deg_count_kernel<<<(E + T - 1) / T, T, 0, stream>>>(dst, dinv, E);
  dinv_kernel<<<(N + T - 1) / T, T, 0, stream>>>(dinv, N);

  const int ntile = N / 16;                       // 100000 is a multiple of 16
  const int gemm_blocks = (ntile + 3) / 4;
  const int nf_blocks = (N * H_DIM + T - 1) / T;
  const int edge_blocks = (int)(((long)E * 16 + T - 1) / T);
  const float inv_n = 1.0f / (float)N;

  auto layer = [&](const float* in, int K, const float* W, const float* b,
                   const float* g, const float* be, const float* a, float* out) {
    gemm_wmma_kernel<<<gemm_blocks, 128, 0, stream>>>(in, W, B0, N, K);
    aggr_init_kernel<<<nf_blocks, T, 0, stream>>>(B0, dinv, b, B1, N);
    aggr_edges_kernel<<<edge_blocks, T, 0, stream>>>(B0, dinv, src, dst, B1, E);
    bn_partial_kernel<<<BN_BLOCKS, 256, 0, stream>>>(B1, partial, N);
    bn_final_kernel<<<1, 128, 0, stream>>>(partial, BN_BLOCKS, g, be, ssbuf, inv_n);
    bn_apply_prelu_kernel<<<nf_blocks, T, 0, stream>>>(B1, ssbuf, a, out, N);
  };

  layer(x,  128, W1, b1, g1, be1, a1, B1);            // y1 -> B1 (in-place over agg)
  layer(B1,  64, W2, b2, g2, be2, a2, B1);            // y2 -> B1
  layer(B1,  64, W3, b3, g3, be3, a3, (float*)d_out); // y3 -> output
}